// PatchSelector_81535659147921
// MI455X (gfx1250) — compile-verified
//
#include <hip/hip_runtime.h>

typedef float v2f __attribute__((ext_vector_type(2)));
typedef float v8f __attribute__((ext_vector_type(8)));

#define C_DIM 64
#define D_DIM 128
#define P_DIM 64
#define TILE_PITCH 65   // pad to avoid 32-way bank conflicts on d-strided reads
#define BOOST_F 1000000.0f

__global__ __launch_bounds__(128)
void PatchSelector_kernel(const float* __restrict__ x,
                          const int*   __restrict__ channel_idx,
                          const float* __restrict__ W,
                          const float* __restrict__ bias,
                          float* __restrict__ out) {
  __shared__ float tile[D_DIM * TILE_PITCH];  // tile[d][p], padded pitch
  __shared__ float skey[P_DIM];               // (score + bias) + BOOST, fp32
  __shared__ int   perm[P_DIM];               // perm[rank] = patch index

  const int b    = blockIdx.x;
  const int tid  = threadIdx.x;
  const int lane = tid & 31;
  const int wave = tid >> 5;

  const int   cid  = channel_idx[0];
  const float bval = bias[0];

  // x[b, cid, d, p] : flat = ((b*C + cid)*D + d)*P + p
  const float* xb = x + (size_t)(b * C_DIM + cid) * (D_DIM * P_DIM);

  // ---- Phase 0: stage the needed 128x64 channel slice into LDS (single HBM read)
  for (int i = tid; i < D_DIM * P_DIM; i += 128) {
    const int d = i >> 6;       // i / P
    const int p = i & (P_DIM - 1);
    tile[d * TILE_PITCH + p] = xb[i];
  }
  __syncthreads();

  // ---- Phase 1: scores via V_WMMA_F32_16X16X4_F32, one 16-patch block per wave
  // A(16x4) rows all broadcast W[4c..4c+3]; B(4x16) = x[4c..4c+3, patchblock].
  // fp32 A layout: lanes 0-15 -> {K=0,K=1}, lanes 16-31 -> {K=2,K=3}; B mirrored.
  {
    const int pb    = wave;                 // patch block 0..3
    const int j     = lane & 15;
    const int khalf = (lane >> 4) << 1;     // 0 for lanes 0-15, 2 for lanes 16-31
    const int jg    = pb * 16 + j;          // global patch column this lane feeds

    v8f acc = {};
    #pragma unroll 4
    for (int c = 0; c < 32; ++c) {          // K = 128 in chunks of 4
      const int d0 = c * 4 + khalf;
      v2f a, bm;
      a.x  = W[d0];
      a.y  = W[d0 + 1];
      bm.x = tile[d0 * TILE_PITCH + jg];
      bm.y = tile[(d0 + 1) * TILE_PITCH + jg];
      // 8 args: (neg_a, A, neg_b, B, c_mod, C, reuse_a, reuse_b)
      acc = __builtin_amdgcn_wmma_f32_16x16x4_f32(
          false, a, false, bm, (short)0, acc, false, false);
    }
    // All C rows are identical; row M=0 lives in acc[0] of lanes 0-15 (N = lane).
    if (lane < 16) {
      const float s = acc[0] + bval;        // einsum + b   (fp32, matches ref)
      skey[jg] = s + BOOST_F;               // + BOOST in fp32 -> tie quantization
    }
  }
  __syncthreads();

  // ---- Phase 2: stable descending rank sort of the 64 keys (O(64^2), parallel)
  if (tid < P_DIM) {
    const float my = skey[tid];
    int rank = 0;
    #pragma unroll 8
    for (int jj = 0; jj < P_DIM; ++jj) {
      const float o = skey[jj];
      rank += (o > my) || (o == my && jj < tid);   // ties -> lower index first
    }
    perm[rank] = tid;                              // ranks are unique
  }
  __syncthreads();

  // ---- Phase 3: gather from LDS, coalesced store: out[b, k, d] = tile[d][perm[k]]
  float* ob = out + (size_t)b * (P_DIM * D_DIM);
  for (int o = tid; o < P_DIM * D_DIM; o += 128) {
    const int k = o >> 7;           // o / D
    const int d = o & (D_DIM - 1);
    ob[o] = tile[d * TILE_PITCH + perm[k]];
  }
}

extern "C" void kernel_launch(void* const* d_in, const int* in_sizes, int n_in,
                              void* d_out, int out_size, void* d_ws, size_t ws_size,
                              hipStream_t stream) {
  const float* x   = (const float*)d_in[0];
  const int*   cid = (const int*)d_in[1];
  const float* W   = (const float*)d_in[2];
  const float* bb  = (const float*)d_in[3];
  float*       out = (float*)d_out;

  const int B = out_size / (P_DIM * D_DIM);   // out is (B, P, D)
  PatchSelector_kernel<<<dim3(B), dim3(128), 0, stream>>>(x, cid, W, bb, out);
}